// _ST_Attention_11098195493496
// MI455X (gfx1250) — compile-verified
//
#include <hip/hip_runtime.h>
#include <hip/hip_bf16.h>
#include <math.h>

#define B_ 8
#define N_ 300
#define T_ 12
#define E_ 512
#define H_ 8
#define D_ 64
#define KT_ 19                     // ceil(300/16) key tiles
#define SCALE_ 0.04419417382415922f // 1/sqrt(512)
#define NEG_ (-1e10f)

typedef __attribute__((ext_vector_type(2))) float v2f;
typedef __attribute__((ext_vector_type(8))) float v8f;

__device__ __forceinline__ v8f wmma_f32(v2f a, v2f b, v8f c) {
    // V_WMMA_F32_16X16X4_F32 : D = A(16x4 f32) x B(4x16 f32) + C(16x16 f32)
    return __builtin_amdgcn_wmma_f32_16x16x4_f32(
        false, a, false, b, (short)0, c, false, false);
}

// ---------------------------------------------------------------------------
// Y[M, 512] = X[M, 512] @ W[512,512]^T + bias   (torch Linear convention)
// one wave -> 16(M) x 64(N) output strip, K looped in steps of 4 via f32 WMMA
// ---------------------------------------------------------------------------
__global__ __launch_bounds__(128) void gemm_bias_kernel(
    const float* __restrict__ X, const float* __restrict__ W,
    const float* __restrict__ bias, float* __restrict__ Y, int M)
{
    const int lane = threadIdx.x & 31;
    const int wave = threadIdx.x >> 5;
    const int half = lane >> 4;   // selects K pair / row half
    const int l16  = lane & 15;

    const int NT = E_ / 64;                       // 8 column groups of 64
    const int tid = blockIdx.x * 4 + wave;
    const int mtiles = M >> 4;
    if (tid >= mtiles * NT) return;
    const int mt = tid / NT;
    const int ng = tid % NT;

    const int m = mt * 16 + l16;                  // A row this lane feeds
    const float* xrow = X + (size_t)m * E_;
    const float* w0 = W + (size_t)(ng * 64 +  0 + l16) * E_;
    const float* w1 = W + (size_t)(ng * 64 + 16 + l16) * E_;
    const float* w2 = W + (size_t)(ng * 64 + 32 + l16) * E_;
    const float* w3 = W + (size_t)(ng * 64 + 48 + l16) * E_;

    v8f c0 = {}, c1 = {}, c2 = {}, c3 = {};
    #pragma unroll 4
    for (int ks = 0; ks < E_ / 4; ++ks) {
        const int k = ks * 4 + half * 2;
        v2f a  = *(const v2f*)(xrow + k);
        v2f b0 = *(const v2f*)(w0 + k);
        v2f b1 = *(const v2f*)(w1 + k);
        v2f b2 = *(const v2f*)(w2 + k);
        v2f b3 = *(const v2f*)(w3 + k);
        c0 = wmma_f32(a, b0, c0);
        c1 = wmma_f32(a, b1, c1);
        c2 = wmma_f32(a, b2, c2);
        c3 = wmma_f32(a, b3, c3);
    }

    #pragma unroll
    for (int nt = 0; nt < 4; ++nt) {
        v8f c = (nt == 0) ? c0 : (nt == 1) ? c1 : (nt == 2) ? c2 : c3;
        const int n = ng * 64 + nt * 16 + l16;
        const float bv = bias[n];
        #pragma unroll
        for (int r = 0; r < 8; ++r) {
            const int row = mt * 16 + r + half * 8;
            Y[(size_t)row * E_ + n] = c[r] + bv;
        }
    }
}

// ---------------------------------------------------------------------------
// Spatial attention over node axis, one (b,t,h, 16-query tile) per wave.
// Online softmax; P staged through per-wave LDS to convert C-layout->A-layout.
// ---------------------------------------------------------------------------
__global__ __launch_bounds__(128) void st_attention_kernel(
    const float* __restrict__ Q, const float* __restrict__ K,
    const float* __restrict__ V, const unsigned char* __restrict__ amask,
    const float* __restrict__ abias, const unsigned char* __restrict__ kmiss,
    float* __restrict__ ctx)
{
    __shared__ float plds[4][16][17];   // per-wave P staging, padded

    const int lane = threadIdx.x & 31;
    const int wave = threadIdx.x >> 5;
    const int half = lane >> 4;
    const int l16  = lane & 15;

    const int bid  = blockIdx.x;
    const int qblk = bid % 5;           // 5 blocks cover 19 q-tiles (4 waves each)
    const int bth  = bid / 5;
    const int h = bth % H_;
    const int t = (bth / H_) % T_;
    const int b = bth / (H_ * T_);
    const int qt = qblk * 4 + wave;
    if (qt >= KT_) return;
    const int qbase = qt * 16;

    // ---- preload Q A-fragments (16 K-steps of 4 over D=64) ----
    const int qrow = qbase + l16;
    const bool qok = (qrow < N_);
    const float* qptr =
        Q + (((size_t)(b * N_ + (qok ? qrow : 0)) * T_ + t) * E_ + h * D_);
    v2f qa[16];
    #pragma unroll
    for (int s = 0; s < 16; ++s) {
        v2f a = {0.f, 0.f};
        if (qok) a = *(const v2f*)(qptr + s * 4 + half * 2);
        qa[s] = a;
    }

    v8f o0 = {}, o1 = {}, o2 = {}, o3 = {};
    float mrow[8], lrow[8];
    #pragma unroll
    for (int r = 0; r < 8; ++r) { mrow[r] = -INFINITY; lrow[r] = 0.f; }

    for (int kt = 0; kt < KT_; ++kt) {
        const int kbase = kt * 16;

        // ---- S tile = Q_tile @ K_tile^T via WMMA ----
        const int krow = kbase + l16;             // key row feeding B frag
        const bool kok = (krow < N_);
        const float* kptr =
            K + (((size_t)(b * N_ + (kok ? krow : 0)) * T_ + t) * E_ + h * D_);
        v8f s8 = {};
        #pragma unroll
        for (int s = 0; s < 16; ++s) {
            v2f bb = {0.f, 0.f};
            if (kok) bb = *(const v2f*)(kptr + s * 4 + half * 2);
            s8 = wmma_f32(qa[s], bb, s8);
        }

        // ---- mask + bias + missing + scale, online softmax update ----
        const int kk = kbase + l16;               // column (key) this lane holds
        #pragma unroll
        for (int r = 0; r < 8; ++r) {
            const int qq = qbase + r + half * 8;
            float sv;
            if (qq < N_ && kk < N_) {
                const size_t mi = (((size_t)t * N_ + qq) * N_ + kk) * H_ + h;
                sv = amask[mi] ? s8[r] : NEG_;
                sv += abias[mi];
                if (kmiss[((size_t)b * N_ + kk) * T_ + t]) sv = NEG_;
                sv *= SCALE_;
            } else {
                sv = (qq < N_) ? -INFINITY : 0.0f;  // pad keys / dead rows
            }
            float mx = sv;
            #pragma unroll
            for (int msk = 1; msk < 16; msk <<= 1)
                mx = fmaxf(mx, __shfl_xor(mx, msk, 32));
            const float mnew  = fmaxf(mrow[r], mx);
            const float alpha = __expf(mrow[r] - mnew);
            const float pe    = __expf(sv - mnew);
            float ssum = pe;
            #pragma unroll
            for (int msk = 1; msk < 16; msk <<= 1)
                ssum += __shfl_xor(ssum, msk, 32);
            lrow[r] = lrow[r] * alpha + ssum;
            mrow[r] = mnew;
            o0[r] *= alpha; o1[r] *= alpha; o2[r] *= alpha; o3[r] *= alpha;
            plds[wave][r + half * 8][l16] = pe;   // C-layout -> LDS
        }

        // ---- O += P @ V_tile  (P re-read from LDS in A-layout) ----
        #pragma unroll
        for (int s = 0; s < 4; ++s) {
            v2f pa;
            pa.x = plds[wave][l16][s * 4 + half * 2];
            pa.y = plds[wave][l16][s * 4 + half * 2 + 1];
            const int k0 = kbase + s * 4 + half * 2;
            const int k1 = k0 + 1;
            const float* vb0 = (k0 < N_)
                ? V + (((size_t)(b * N_ + k0) * T_ + t) * E_ + h * D_) : nullptr;
            const float* vb1 = (k1 < N_)
                ? V + (((size_t)(b * N_ + k1) * T_ + t) * E_ + h * D_) : nullptr;
            #pragma unroll
            for (int dt = 0; dt < 4; ++dt) {
                const int d = dt * 16 + l16;
                v2f vv;
                vv.x = vb0 ? vb0[d] : 0.f;
                vv.y = vb1 ? vb1[d] : 0.f;
                if (dt == 0) o0 = wmma_f32(pa, vv, o0);
                else if (dt == 1) o1 = wmma_f32(pa, vv, o1);
                else if (dt == 2) o2 = wmma_f32(pa, vv, o2);
                else o3 = wmma_f32(pa, vv, o3);
            }
        }
    }

    // ---- normalize + NaN scrub + store ctx [B,N,T,E] ----
    #pragma unroll
    for (int r = 0; r < 8; ++r) {
        const int qq = qbase + r + half * 8;
        if (qq >= N_) continue;
        const float inv = 1.0f / lrow[r];
        const size_t base = (((size_t)(b * N_ + qq) * T_ + t) * E_ + h * D_);
        float x0 = o0[r] * inv; if (x0 != x0) x0 = 0.f;
        float x1 = o1[r] * inv; if (x1 != x1) x1 = 0.f;
        float x2 = o2[r] * inv; if (x2 != x2) x2 = 0.f;
        float x3 = o3[r] * inv; if (x3 != x3) x3 = 0.f;
        ctx[base +  0 + l16] = x0;
        ctx[base + 16 + l16] = x1;
        ctx[base + 32 + l16] = x2;
        ctx[base + 48 + l16] = x3;
    }
}

// ---------------------------------------------------------------------------
extern "C" void kernel_launch(void* const* d_in, const int* in_sizes, int n_in,
                              void* d_out, int out_size, void* d_ws, size_t ws_size,
                              hipStream_t stream)
{
    const float* value = (const float*)d_in[0];
    const float* key   = (const float*)d_in[1];
    const float* query = (const float*)d_in[2];
    const unsigned char* amask = (const unsigned char*)d_in[3];
    const float* abias = (const float*)d_in[4];
    const unsigned char* kmiss = (const unsigned char*)d_in[5];
    const float* Wv = (const float*)d_in[6];  const float* bv = (const float*)d_in[7];
    const float* Wk = (const float*)d_in[8];  const float* bk = (const float*)d_in[9];
    const float* Wq = (const float*)d_in[10]; const float* bq = (const float*)d_in[11];
    const float* Wo = (const float*)d_in[12]; const float* bo = (const float*)d_in[13];
    float* out = (float*)d_out;

    const size_t S = (size_t)B_ * N_ * T_ * E_;   // 14,745,600 floats
    float* qw = (float*)d_ws;
    float* kw = qw + S;
    float* vw = kw + S;
    float* cw = vw + S;

    const int M = B_ * N_ * T_;                   // 28800 rows
    const int waves = (M / 16) * (E_ / 64);       // 14400 waves
    dim3 blk(128);
    dim3 grd(waves / 4);                          // 3600 blocks

    hipLaunchKernelGGL(gemm_bias_kernel, grd, blk, 0, stream, value, Wv, bv, vw, M);
    hipLaunchKernelGGL(gemm_bias_kernel, grd, blk, 0, stream, key,   Wk, bk, kw, M);
    hipLaunchKernelGGL(gemm_bias_kernel, grd, blk, 0, stream, query, Wq, bq, qw, M);

    dim3 agrd(B_ * T_ * H_ * 5);                  // 3840 blocks, 4 waves each
    hipLaunchKernelGGL(st_attention_kernel, agrd, blk, 0, stream,
                       qw, kw, vw, amask, abias, kmiss, cw);

    hipLaunchKernelGGL(gemm_bias_kernel, grd, blk, 0, stream, cw, Wo, bo, out, M);
}